// NTM_5660766896261
// MI455X (gfx1250) — compile-verified
//
#include <hip/hip_runtime.h>
#include <math.h>

// Problem constants (from reference setup_inputs)
#define BB 128
#define TT 128
#define DD 128
#define NN 128
#define MM 128
#define HH 256
#define BQ 16          // batch rows per workgroup (WMMA tile height)
#define NTHREADS 512   // 16 waves of 32
#define EPSV 1e-8f

typedef float v2f __attribute__((ext_vector_type(2)));
typedef float v8f __attribute__((ext_vector_type(8)));

__device__ __forceinline__ float sigmoidf_(float x) { return 1.f / (1.f + __expf(-x)); }
__device__ __forceinline__ float softplusf_(float x) { return (x > 30.f) ? x : log1pf(__expf(x)); }

__device__ __forceinline__ float wave_sum(float v) {
#pragma unroll
  for (int off = 16; off > 0; off >>= 1) v += __shfl_xor(v, off, 32);
  return v;
}
__device__ __forceinline__ float wave_max(float v) {
#pragma unroll
  for (int off = 16; off > 0; off >>= 1) v = fmaxf(v, __shfl_xor(v, off, 32));
  return v;
}

// C[16x16] += A[16xK] * W[K x ldw] (column tile at ncol), fp32 WMMA 16x16x4.
// A layout (ISA 7.12.2, 32-bit A 16x4): lanes 0-15 hold K=k,k+1; lanes 16-31 hold K=k+2,k+3.
// B layout mirrored: lane holds one column, VGPRs hold K values.
__device__ __forceinline__ v8f wmma_acc(const float* __restrict__ A, int lda,
                                        const float* __restrict__ W, int ldw,
                                        int ncol, int K, v8f c) {
  const int lane = threadIdx.x & 31;
  const int hlf  = lane >> 4;
  const int l    = lane & 15;
#pragma unroll 4
  for (int k = 0; k < K; k += 4) {
    v2f a, b;
    a.x = A[l * lda + k + 2 * hlf];
    a.y = A[l * lda + k + 2 * hlf + 1];
    b.x = W[(size_t)(k + 2 * hlf) * ldw + ncol + l];
    b.y = W[(size_t)(k + 2 * hlf + 1) * ldw + ncol + l];
    c = __builtin_amdgcn_wmma_f32_16x16x4_f32(false, a, false, b, (short)0, c, false, false);
  }
  return c;
}

__global__ __launch_bounds__(NTHREADS, 1) void ntm_persistent_kernel(
    const float* __restrict__ x, const float* __restrict__ mem0,
    const float* __restrict__ wr0, const float* __restrict__ ww0,
    const float* __restrict__ h0, const float* __restrict__ Wx,
    const float* __restrict__ Wrd, const float* __restrict__ bh,
    const float* __restrict__ Wk, const float* __restrict__ bk,
    const float* __restrict__ Wb, const float* __restrict__ bb,
    const float* __restrict__ Wg, const float* __restrict__ bg,
    const float* __restrict__ Ws, const float* __restrict__ bs,
    const float* __restrict__ Wgam, const float* __restrict__ bgam,
    const float* __restrict__ We, const float* __restrict__ be,
    const float* __restrict__ Wa, const float* __restrict__ ba,
    float* __restrict__ out, float* __restrict__ memws) {
  __shared__ float h_lds[BQ][HH];        // controller state
  __shared__ float r_lds[BQ][MM];        // read vector
  __shared__ float er_lds[BQ][MM];       // erase
  __shared__ float ad_lds[BQ][MM];       // add
  __shared__ float k_lds[2][BQ][MM];     // keys per head
  __shared__ float w_lds[2][BQ][NN];     // [0]=wr, [1]=ww
  __shared__ float dot_lds[2][BQ][NN];   // content dot per head
  __shared__ float nrm_lds[BQ][NN];      // mem row norms
  __shared__ float knorm_s[2][BQ];
  __shared__ float beta_s[2][BQ], g_s[2][BQ], gam_s[2][BQ], s_s[2][BQ][3];

  const int tid  = threadIdx.x;
  const int wave = tid >> 5;
  const int lane = tid & 31;
  const int hlf  = lane >> 4;
  const int l    = lane & 15;
  const int b0   = blockIdx.x * BQ;

  // --- load initial state into LDS ---
  for (int i = tid; i < BQ * HH; i += NTHREADS)
    h_lds[i / HH][i % HH] = h0[(size_t)(b0 + i / HH) * HH + (i % HH)];
  for (int i = tid; i < BQ * NN; i += NTHREADS) {
    w_lds[0][i / NN][i % NN] = wr0[(size_t)(b0 + i / NN) * NN + (i % NN)];
    w_lds[1][i / NN][i % NN] = ww0[(size_t)(b0 + i / NN) * NN + (i % NN)];
  }
  __syncthreads();

  for (int t = 0; t < TT; ++t) {
    // ---------- Phase A: erase = sigmoid(h We + be), add = tanh(h Wa + ba) ----------
    {
      const int nc = (wave & 7) * 16;
      const float* W = (wave < 8) ? We : Wa;
      v8f c = {0.f, 0.f, 0.f, 0.f, 0.f, 0.f, 0.f, 0.f};
      c = wmma_acc(&h_lds[0][0], HH, W, MM, nc, HH, c);
      if (wave < 8) {
#pragma unroll
        for (int i = 0; i < 8; ++i)
          er_lds[i + 8 * hlf][nc + l] = sigmoidf_(c[i] + be[nc + l]);
      } else {
#pragma unroll
        for (int i = 0; i < 8; ++i)
          ad_lds[i + 8 * hlf][nc + l] = tanhf(c[i] + ba[nc + l]);
      }
    }
    __syncthreads();

    // ---------- Phase B: mem = mem*(1 - ww*erase) + ww*add ; r = wr . mem ----------
    {
      const int m  = tid & (MM - 1);
      const int bq = tid >> 7;  // 0..3
#pragma unroll
      for (int bi = 0; bi < 4; ++bi) {
        const int b = bq + 4 * bi;
        const float er = er_lds[b][m];
        const float ad = ad_lds[b][m];
        float racc = 0.f;
        float* mrow = memws + ((size_t)(b0 + b) * NN) * MM + m;
        const float* m0row = mem0 + m;
        for (int n = 0; n < NN; ++n) {
          float v = (t == 0) ? m0row[(size_t)n * MM] : mrow[(size_t)n * MM];
          const float wwv = w_lds[1][b][n];
          v = v * (1.f - wwv * er) + wwv * ad;
          mrow[(size_t)n * MM] = v;
          racc += w_lds[0][b][n] * v;
        }
        r_lds[b][m] = racc;
      }
    }
    __threadfence_block();
    __syncthreads();

    // ---------- Phase D: h = tanh(x_t Wx + r Wrd + bh) ; emit output ----------
    {
      const int nc = wave * 16;
      v8f c = {0.f, 0.f, 0.f, 0.f, 0.f, 0.f, 0.f, 0.f};
      c = wmma_acc(x + (size_t)b0 * TT * DD + (size_t)t * DD, TT * DD, Wx, HH, nc, DD, c);
      c = wmma_acc(&r_lds[0][0], MM, Wrd, HH, nc, MM, c);
#pragma unroll
      for (int i = 0; i < 8; ++i) {
        const int row = i + 8 * hlf;
        const float val = tanhf(c[i] + bh[nc + l]);
        h_lds[row][nc + l] = val;
        out[((size_t)(b0 + row) * TT + t) * HH + nc + l] = val;
      }
    }
    __syncthreads();

    // ---------- Phase E: keys k = tanh(h Wk + bk) + scalar head projections ----------
    {
      const int hd = wave >> 3;
      const int nc = (wave & 7) * 16;
      v8f c = {0.f, 0.f, 0.f, 0.f, 0.f, 0.f, 0.f, 0.f};
      c = wmma_acc(&h_lds[0][0], HH, Wk + (size_t)hd * HH * MM, MM, nc, HH, c);
#pragma unroll
      for (int i = 0; i < 8; ++i)
        k_lds[hd][i + 8 * hlf][nc + l] = tanhf(c[i] + bk[hd * MM + nc + l]);

      if (tid < 192) {  // waves 0..5: 2 heads * 16 b * 6 projections
        const int hd2 = tid / 96;
        const int rem = tid % 96;
        const int b = rem / 6;
        const int p = rem % 6;
        float acc = 0.f;
        if (p < 3) {
          const float* wv = ((p == 0) ? Wb : (p == 1) ? Wg : Wgam) + hd2 * HH;
          for (int j = 0; j < HH; ++j) acc += h_lds[b][j] * wv[j];
          acc += ((p == 0) ? bb : (p == 1) ? bg : bgam)[hd2];
          if (p == 0)      beta_s[hd2][b] = softplusf_(acc);
          else if (p == 1) g_s[hd2][b]    = sigmoidf_(acc);
          else             gam_s[hd2][b]  = 1.f + softplusf_(acc);
        } else {
          const int comp = p - 3;
          const float* wsv = Ws + (size_t)hd2 * HH * 3 + comp;
          for (int j = 0; j < HH; ++j) acc += h_lds[b][j] * wsv[j * 3];
          s_s[hd2][b][comp] = acc + bs[hd2 * 3 + comp];  // softmax applied in F
        }
      }
    }
    __syncthreads();

    // ---------- Phase C: ||k||, content dots, ||mem rows|| ----------
    {
#pragma unroll
      for (int it = 0; it < 2; ++it) {
        const int task = wave + 16 * it;
        const int hd = task >> 4, b = task & 15;
        float acc = 0.f;
#pragma unroll
        for (int cc = 0; cc < 4; ++cc) {
          const float v = k_lds[hd][b][lane + 32 * cc];
          acc += v * v;
        }
        acc = wave_sum(acc);
        if (lane == 0) knorm_s[hd][b] = sqrtf(acc);
      }
      for (int i = 0; i < 128; ++i) {
        const int row = wave + 16 * i;   // 2048 (b,n) rows over 16 waves
        const int b = row >> 7, n = row & 127;
        const float* mrow = memws + ((size_t)(b0 + b) * NN + n) * MM;
        float an = 0.f, a0 = 0.f, a1 = 0.f;
#pragma unroll
        for (int cc = 0; cc < 4; ++cc) {
          const int m = lane + 32 * cc;
          const float v = mrow[m];
          an += v * v;
          a0 += k_lds[0][b][m] * v;
          a1 += k_lds[1][b][m] * v;
        }
        an = wave_sum(an); a0 = wave_sum(a0); a1 = wave_sum(a1);
        if (lane == 0) {
          nrm_lds[b][n] = sqrtf(an);
          dot_lds[0][b][n] = a0;
          dot_lds[1][b][n] = a1;
        }
      }
    }
    __syncthreads();

    // ---------- Phase F: softmax, gate, circular shift, sharpen, renorm ----------
#pragma unroll
    for (int it = 0; it < 2; ++it) {
      const int task = wave + 16 * it;
      const int hd = task >> 4, b = task & 15;
      const float beta = beta_s[hd][b], g = g_s[hd][b], gamma = gam_s[hd][b];
      const float kn = knorm_s[hd][b];
      // 3-tap shift softmax (redundant per lane)
      const float sr0 = s_s[hd][b][0], sr1 = s_s[hd][b][1], sr2 = s_s[hd][b][2];
      const float smax = fmaxf(sr0, fmaxf(sr1, sr2));
      const float e0 = __expf(sr0 - smax), e1 = __expf(sr1 - smax), e2 = __expf(sr2 - smax);
      const float sinv = 1.f / (e0 + e1 + e2);
      const float s0 = e0 * sinv, s1 = e1 * sinv, s2 = e2 * sinv;

      float z[4];
#pragma unroll
      for (int cc = 0; cc < 4; ++cc) {
        const int n = lane + 32 * cc;
        const float sim = dot_lds[hd][b][n] / (kn * nrm_lds[b][n] + EPSV);
        z[cc] = beta * sim;
      }
      float mx = fmaxf(fmaxf(z[0], z[1]), fmaxf(z[2], z[3]));
      mx = wave_max(mx);
      float e[4], sum = 0.f;
#pragma unroll
      for (int cc = 0; cc < 4; ++cc) { e[cc] = __expf(z[cc] - mx); sum += e[cc]; }
      sum = wave_sum(sum);
      float wgv[4];
#pragma unroll
      for (int cc = 0; cc < 4; ++cc) {
        const int n = lane + 32 * cc;
        wgv[cc] = g * (e[cc] / sum) + (1.f - g) * w_lds[hd][b][n];
      }
      // circular 3-tap conv: n indexed as lane + 32*chunk, neighbors via lane shuffles
      const int lm1 = (lane + 31) & 31, lp1 = (lane + 1) & 31;
      float sh[4];
#pragma unroll
      for (int cc = 0; cc < 4; ++cc) {
        const float a1v = __shfl(wgv[cc], lm1, 32);
        const float a2v = __shfl(wgv[(cc + 3) & 3], lm1, 32);
        const float dn = (lane == 0) ? a2v : a1v;   // value at n-1 (circular)
        const float b1v = __shfl(wgv[cc], lp1, 32);
        const float b2v = __shfl(wgv[(cc + 1) & 3], lp1, 32);
        const float up = (lane == 31) ? b2v : b1v;  // value at n+1 (circular)
        sh[cc] = s0 * dn + s1 * wgv[cc] + s2 * up;
      }
      float wp[4], wsum = 0.f;
#pragma unroll
      for (int cc = 0; cc < 4; ++cc) { wp[cc] = powf(sh[cc], gamma); wsum += wp[cc]; }
      wsum = wave_sum(wsum);
      const float inv = 1.f / (wsum + EPSV);
#pragma unroll
      for (int cc = 0; cc < 4; ++cc) w_lds[hd][b][lane + 32 * cc] = wp[cc] * inv;
    }
    __syncthreads();
  }
}

extern "C" void kernel_launch(void* const* d_in, const int* in_sizes, int n_in,
                              void* d_out, int out_size, void* d_ws, size_t ws_size,
                              hipStream_t stream) {
  (void)in_sizes; (void)n_in; (void)out_size; (void)ws_size;
  const float* x    = (const float*)d_in[0];
  const float* mem0 = (const float*)d_in[1];
  const float* wr0  = (const float*)d_in[2];
  const float* ww0  = (const float*)d_in[3];
  const float* h0   = (const float*)d_in[4];
  const float* Wx   = (const float*)d_in[5];
  const float* Wrd  = (const float*)d_in[6];
  const float* bh   = (const float*)d_in[7];
  const float* Wk   = (const float*)d_in[8];
  const float* bk   = (const float*)d_in[9];
  const float* Wb   = (const float*)d_in[10];
  const float* bb   = (const float*)d_in[11];
  const float* Wg   = (const float*)d_in[12];
  const float* bg   = (const float*)d_in[13];
  const float* Ws   = (const float*)d_in[14];
  const float* bs   = (const float*)d_in[15];
  const float* Wgam = (const float*)d_in[16];
  const float* bgam = (const float*)d_in[17];
  const float* We   = (const float*)d_in[18];
  const float* be   = (const float*)d_in[19];
  const float* Wa   = (const float*)d_in[20];
  const float* ba   = (const float*)d_in[21];

  dim3 grid(BB / BQ);        // 8 persistent workgroups
  dim3 block(NTHREADS);      // 16 waves
  ntm_persistent_kernel<<<grid, block, 0, stream>>>(
      x, mem0, wr0, ww0, h0, Wx, Wrd, bh, Wk, bk, Wb, bb, Wg, bg, Ws, bs,
      Wgam, bgam, We, be, Wa, ba, (float*)d_out, (float*)d_ws);
}